// ClassicGNN_21844203667598
// MI455X (gfx1250) — compile-verified
//
#include <hip/hip_runtime.h>
#include <hip/hip_bf16.h>

typedef __attribute__((ext_vector_type(2))) float v2f;
typedef __attribute__((ext_vector_type(8))) float v8f;

#define WAVE 32

// ---------------------------------------------------------------------------
// Degree count: one thread per edge, fp32 atomic add into deg[dst].
// ---------------------------------------------------------------------------
__global__ __launch_bounds__(256) void gnn_deg_count(const int* __restrict__ dst,
                                                     float* __restrict__ deg, int E) {
    int e = blockIdx.x * blockDim.x + threadIdx.x;
    if (e >= E) return;
    __hip_atomic_fetch_add(deg + dst[e], 1.0f, __ATOMIC_RELAXED, __HIP_MEMORY_SCOPE_AGENT);
}

// ---------------------------------------------------------------------------
// Edge scatter-add: agg[dst] += feat[src]. One thread per float4 chunk.
// D templated (128/256) so the chunk index math is shifts, not divides.
// ---------------------------------------------------------------------------
template <int D>
__global__ __launch_bounds__(256) void gnn_scatter_add(const float* __restrict__ feat,
                                                       const int* __restrict__ src,
                                                       const int* __restrict__ dst,
                                                       float* __restrict__ agg, int E) {
    constexpr int DQ = D >> 2;             // float4 chunks per row
    long long tid = (long long)blockIdx.x * blockDim.x + threadIdx.x;
    if (tid >= (long long)E * DQ) return;
    int e = (int)(tid / DQ);
    int q = (int)(tid % DQ);
    long long s = src[e];
    long long d = dst[e];
    const float4 v = reinterpret_cast<const float4*>(feat + s * (long long)D)[q];
    float* base = agg + d * (long long)D + ((long long)q << 2);
    __hip_atomic_fetch_add(base + 0, v.x, __ATOMIC_RELAXED, __HIP_MEMORY_SCOPE_AGENT);
    __hip_atomic_fetch_add(base + 1, v.y, __ATOMIC_RELAXED, __HIP_MEMORY_SCOPE_AGENT);
    __hip_atomic_fetch_add(base + 2, v.z, __ATOMIC_RELAXED, __HIP_MEMORY_SCOPE_AGENT);
    __hip_atomic_fetch_add(base + 3, v.w, __ATOMIC_RELAXED, __HIP_MEMORY_SCOPE_AGENT);
}

// ---------------------------------------------------------------------------
// Fused SAGE layer: out = relu( (agg/deg) @ Wl + b + x @ Wr ), DOUT = 256.
// One wave owns a 16x64 output tile = 4 x (16x16) accumulators; FP32 WMMA
// 16x16x4. Per k-step: one b64 A load + pk_mul (deg-normalize), all 4 B
// fragments gathered first (single load clause), then 4 back-to-back WMMAs.
//   A fragment (16x4 f32): lane l -> row (l&15), K = 2*(l>>4) + {0,1}
//   B fragment (4x16 f32): lane l -> col (l&15), K = 2*(l>>4) + {0,1}
//   C/D (16x16 f32):       VGPR v, lane l -> row v + 8*(l>>4), col (l&15)
// N=50000 and DOUT=256 tile exactly -> no lane divergence, EXEC all-ones.
// ---------------------------------------------------------------------------
template <int K>
__global__ __launch_bounds__(256) void gnn_sage_wmma(const float* __restrict__ agg,
                                                     const float* __restrict__ x,
                                                     const float* __restrict__ Wl,
                                                     const float* __restrict__ Wr,
                                                     const float* __restrict__ bias,
                                                     const float* __restrict__ deg,
                                                     float* __restrict__ out, int N) {
    constexpr int DOUT = 256;
    constexpr int NSUB = 4;                       // 4 n-subtiles -> 64 cols/wave
    constexpr int NGRP = DOUT >> 6;               // 64-col groups = 4
    int wave   = (blockIdx.x * blockDim.x + threadIdx.x) >> 5;
    int lane   = threadIdx.x & (WAVE - 1);
    int tilesM = (N + 15) >> 4;
    int tm = wave >> 2;                           // wave / NGRP
    int tg = wave & (NGRP - 1);
    if (tm >= tilesM) return;                     // wave-uniform: EXEC stays full

    int mbase = tm << 4;
    int nbase = tg << 6;
    int mrow  = mbase + (lane & 15);              // A-row handled by this lane
    int ncol0 = nbase + (lane & 15);              // first B/D col for this lane
    int koff  = (lane >> 4) << 1;                 // K offset within 4-step: 0 or 2

    float invd = 1.0f / fmaxf(deg[mrow], 1.0f);

    v8f c[NSUB];
#pragma unroll
    for (int j = 0; j < NSUB; ++j) {
        float bv = bias[ncol0 + 16 * j];
#pragma unroll
        for (int i = 0; i < 8; ++i) c[j][i] = bv;
    }

    const float* arow = agg + (long long)mrow * K + koff;
    const float* xrow = x   + (long long)mrow * K + koff;
    const float* bl   = Wl  + (long long)koff * DOUT + ncol0;
    const float* br   = Wr  + (long long)koff * DOUT + ncol0;

    // (agg / deg) @ Wl  -- normalization fused into the shared A fragment
#pragma unroll 2
    for (int k = 0; k < K; k += 4) {
        v2f a = *(const v2f*)(arow + k);
        a.x *= invd;
        a.y *= invd;
        v2f b[NSUB];
#pragma unroll
        for (int j = 0; j < NSUB; ++j) {          // gather all B frags first
            b[j].x = bl[16 * j];
            b[j].y = bl[DOUT + 16 * j];
        }
        bl += 4 * DOUT;
#pragma unroll
        for (int j = 0; j < NSUB; ++j)
            c[j] = __builtin_amdgcn_wmma_f32_16x16x4_f32(false, a, false, b[j], (short)0,
                                                         c[j], false, false);
    }
    // x @ Wr
#pragma unroll 2
    for (int k = 0; k < K; k += 4) {
        v2f a = *(const v2f*)(xrow + k);
        v2f b[NSUB];
#pragma unroll
        for (int j = 0; j < NSUB; ++j) {
            b[j].x = br[16 * j];
            b[j].y = br[DOUT + 16 * j];
        }
        br += 4 * DOUT;
#pragma unroll
        for (int j = 0; j < NSUB; ++j)
            c[j] = __builtin_amdgcn_wmma_f32_16x16x4_f32(false, a, false, b[j], (short)0,
                                                         c[j], false, false);
    }

    int rbase = mbase + ((lane >> 4) << 3);       // +8 rows for upper lane half
#pragma unroll
    for (int j = 0; j < NSUB; ++j) {
#pragma unroll
        for (int v = 0; v < 8; ++v) {
            out[(long long)(rbase + v) * DOUT + ncol0 + 16 * j] = fmaxf(c[j][v], 0.0f);
        }
    }
}

// ---------------------------------------------------------------------------
// Final layer (Dout = 2) fused with softmax: one wave per node row, dual GEMV
// over K=256 with wave32 butterfly reduction, then stable 2-way softmax.
// ---------------------------------------------------------------------------
__global__ __launch_bounds__(256) void gnn_out_softmax(const float* __restrict__ agg,
                                                       const float* __restrict__ h,
                                                       const float* __restrict__ W3l,
                                                       const float* __restrict__ W3r,
                                                       const float* __restrict__ b3,
                                                       const float* __restrict__ deg,
                                                       float* __restrict__ out,
                                                       int N, int K) {
    int wave = (blockIdx.x * blockDim.x + threadIdx.x) >> 5;
    int lane = threadIdx.x & (WAVE - 1);
    if (wave >= N) return;

    const float* arow = agg + (long long)wave * K;
    const float* hrow = h   + (long long)wave * K;
    float invd = 1.0f / fmaxf(deg[wave], 1.0f);

    float s0 = 0.0f, s1 = 0.0f;
    for (int k = lane; k < K; k += WAVE) {
        float av = arow[k] * invd;
        float hv = hrow[k];
        s0 += av * W3l[2 * k]     + hv * W3r[2 * k];
        s1 += av * W3l[2 * k + 1] + hv * W3r[2 * k + 1];
    }
#pragma unroll
    for (int off = 16; off > 0; off >>= 1) {
        s0 += __shfl_xor(s0, off, WAVE);
        s1 += __shfl_xor(s1, off, WAVE);
    }
    if (lane == 0) {
        float y0 = s0 + b3[0];
        float y1 = s1 + b3[1];
        float m  = fmaxf(y0, y1);
        float e0 = __expf(y0 - m);
        float e1 = __expf(y1 - m);
        float inv = 1.0f / (e0 + e1);
        out[2 * (long long)wave + 0] = e0 * inv;
        out[2 * (long long)wave + 1] = e1 * inv;
    }
}

// ---------------------------------------------------------------------------
extern "C" void kernel_launch(void* const* d_in, const int* in_sizes, int n_in,
                              void* d_out, int out_size, void* d_ws, size_t ws_size,
                              hipStream_t stream) {
    constexpr int D_IN = 128, D_HID = 256;

    const float* x   = (const float*)d_in[0];
    const int*   ei  = (const int*)d_in[1];
    const float* W1l = (const float*)d_in[2];
    const float* b1  = (const float*)d_in[3];
    const float* W1r = (const float*)d_in[4];
    const float* W2l = (const float*)d_in[5];
    const float* b2  = (const float*)d_in[6];
    const float* W2r = (const float*)d_in[7];
    const float* W3l = (const float*)d_in[8];
    const float* b3  = (const float*)d_in[9];
    const float* W3r = (const float*)d_in[10];

    const int N = in_sizes[0] / D_IN;      // 50000
    const int E = in_sizes[1] / 2;         // 600000
    const int* srcI = ei;
    const int* dstI = ei + E;

    // Workspace layout (floats): deg | AGG (N x 256, reused) | H1 | H2
    float*  wsf = (float*)d_ws;
    size_t  off = 0;
    float* deg = wsf + off;  off += ((size_t)N + 63) & ~(size_t)63;
    float* AGG = wsf + off;  off += (size_t)N * D_HID;
    float* H1  = wsf + off;  off += (size_t)N * D_HID;
    float* H2  = wsf + off;  off += (size_t)N * D_HID;
    float* OUT = (float*)d_out;

    const int T = 256;
    auto cdiv = [](long long a, long long b) { return (int)((a + b - 1) / b); };

    const int gemmWaves = ((N + 15) / 16) * (D_HID / 64);  // 3125 * 4 = 12500
    const int gemmGrid  = cdiv(gemmWaves, 8);              // 8 waves / block

    // Degrees (shared by all three layers)
    hipMemsetAsync(deg, 0, (size_t)N * sizeof(float), stream);
    gnn_deg_count<<<cdiv(E, T), T, 0, stream>>>(dstI, deg, E);

    // ---- Layer 1: x[128] -> H1[256], ReLU
    hipMemsetAsync(AGG, 0, (size_t)N * D_IN * sizeof(float), stream);
    gnn_scatter_add<D_IN><<<cdiv((long long)E * (D_IN / 4), T), T, 0, stream>>>(x, srcI, dstI, AGG, E);
    gnn_sage_wmma<D_IN><<<gemmGrid, T, 0, stream>>>(AGG, x, W1l, W1r, b1, deg, H1, N);

    // ---- Layer 2: H1[256] -> H2[256], ReLU
    hipMemsetAsync(AGG, 0, (size_t)N * D_HID * sizeof(float), stream);
    gnn_scatter_add<D_HID><<<cdiv((long long)E * (D_HID / 4), T), T, 0, stream>>>(H1, srcI, dstI, AGG, E);
    gnn_sage_wmma<D_HID><<<gemmGrid, T, 0, stream>>>(AGG, H1, W2l, W2r, b2, deg, H2, N);

    // ---- Layer 3: H2[256] -> out[2], softmax
    hipMemsetAsync(AGG, 0, (size_t)N * D_HID * sizeof(float), stream);
    gnn_scatter_add<D_HID><<<cdiv((long long)E * (D_HID / 4), T), T, 0, stream>>>(H2, srcI, dstI, AGG, E);
    gnn_out_softmax<<<cdiv((long long)N * 32, T), T, 0, stream>>>(AGG, H2, W3l, W3r, b3, deg, OUT, N, D_HID);
}